// MPL_77902116815083
// MI455X (gfx1250) — compile-verified
//
#include <hip/hip_runtime.h>
#include <hip/hip_bf16.h>

// ---------------------------------------------------------------------------
// MI455X (gfx1250) kernel: transcendental-bound triangular reduction.
// Strategy: precompute per-k terms once; main kernel = 1 row per wave32,
// k striped over lanes, k-tiles double-buffered in LDS via async global->LDS
// loads (shared by 8 waves/block); fixed-order deterministic final reduce.
// Hot loop uses raw v_log_f32/v_exp_f32 (range-analyzed: exp2 argument in
// [-5, 0] for live terms, so libm's denorm-guard sequence is dead weight).
// ---------------------------------------------------------------------------

#define TPB   256          // threads per block (8 wave32 waves)
#define RPB   8            // rows per block (one per wave)
#define TILE  512          // k elements staged per LDS buffer
#define HUB_DELTA 0.001f

typedef __attribute__((address_space(1))) int gbl_int;   // '__device__ int *'
typedef __attribute__((address_space(3))) int lds_int;   // '__shared__ int *'

#if defined(__HIP_DEVICE_COMPILE__) && __has_builtin(__builtin_amdgcn_global_load_async_to_lds_b32)
#define USE_ASYNC 1
#warning "CDNA5 path: global_load_async_to_lds builtins available"
#else
#define USE_ASYNC 0
#if defined(__HIP_DEVICE_COMPILE__)
#warning "CDNA5 path: async builtins NOT available -> sync LDS staging fallback"
#endif
#endif

// raw v_exp_f32 (no libm denorm/overflow fixup) for the range-safe hot loop
__device__ __forceinline__ float fast_exp2(float x) {
#if defined(__HIP_DEVICE_COMPILE__) && __has_builtin(__builtin_amdgcn_exp2f)
    return __builtin_amdgcn_exp2f(x);
#else
    return exp2f(x);
#endif
}

#if USE_ASYNC
__device__ __forceinline__ void async_ld_b32(const float* gp, float* lp) {
    __builtin_amdgcn_global_load_async_to_lds_b32(
        (gbl_int*)(unsigned long long)(const void*)gp,          // global src (AS1)
        (lds_int*)(unsigned)(unsigned long long)(void*)lp,      // LDS dst (AS3, low 32 bits)
        /*offset=*/0, /*cpol=*/0);
}
__device__ __forceinline__ void wait_async_le6() {
#if __has_builtin(__builtin_amdgcn_s_wait_asynccnt)
    __builtin_amdgcn_s_wait_asynccnt(6);
#else
    asm volatile("s_wait_asynccnt 0x6" ::: "memory");
#endif
}
__device__ __forceinline__ void wait_async_le0() {
#if __has_builtin(__builtin_amdgcn_s_wait_asynccnt)
    __builtin_amdgcn_s_wait_asynccnt(0);
#else
    asm volatile("s_wait_asynccnt 0x0" ::: "memory");
#endif
}
#endif

// ---- Kernel 1: per-k precompute: c[k] = C*lrs[k]^-gamma, P[k] = lr_sum[k-1]
__global__ __launch_bounds__(256)
void mpl_precompute(const float* __restrict__ lrs, const float* __restrict__ lr_sum,
                    const float* pC, const float* pGamma,
                    float* __restrict__ cArr, float* __restrict__ pArr, int T) {
    int k = blockIdx.x * blockDim.x + threadIdx.x;
    if (k >= T) return;
    float lr = lrs[k];
    cArr[k] = pC[0] * exp2f(-pGamma[0] * __log2f(lr));     // runs T times only
    pArr[k] = lr_sum[k] - lr;                              // == lr_sum[k-1], 0 at k==0
}

// ---- Kernel 2: main triangular sum; one row per wave32, tiles in LDS
__global__ __launch_bounds__(TPB)
void mpl_main_kernel(const float* __restrict__ S1, const float* __restrict__ lr_gap,
                     const int* __restrict__ step, const float* __restrict__ loss,
                     const float* __restrict__ cArr, const float* __restrict__ pArr,
                     const float* pL0, const float* pA, const float* pAlpha,
                     const float* pB, const float* pBeta,
                     float* __restrict__ rowH, int M, int T) {
    __shared__ __align__(16) float sC[2][TILE];
    __shared__ __align__(16) float sP[2][TILE];
    __shared__ __align__(16) float sG[2][TILE];
    __shared__ int sStep[RPB];

    const int tid  = threadIdx.x;
    const int lane = tid & 31;
    const int wid  = tid >> 5;
    const int s    = blockIdx.x * RPB + wid;
    const float beta = pBeta[0];

    int   step_s = -1;
    float S1s    = 1.0f;
    if (s < M) { step_s = step[s]; S1s = S1[s]; }
    if (lane == 0) sStep[wid] = step_s;
    __syncthreads();
    int kmax = 0;
#pragma unroll
    for (int i = 0; i < RPB; ++i) kmax = max(kmax, sStep[i]);
    const int ntiles = (kmax + TILE) / TILE;   // ceil((kmax+1)/TILE)

    // cooperative tile stage: 256 threads x 2 elems x 3 arrays (6 b32s/thread)
    auto issue_tile = [&](int t, int b) {
        const int base = t * TILE;
#pragma unroll
        for (int e = 0; e < TILE / TPB; ++e) {
            const int kl = e * TPB + tid;
            const int kg = min(base + kl, T - 1);   // clamp -> uniform issue count
#if USE_ASYNC
            async_ld_b32(cArr   + kg, &sC[b][kl]);
            async_ld_b32(pArr   + kg, &sP[b][kl]);
            async_ld_b32(lr_gap + kg, &sG[b][kl]);
#else
            sC[b][kl] = cArr[kg];
            sP[b][kl] = pArr[kg];
            sG[b][kl] = lr_gap[kg];
#endif
        }
    };

    issue_tile(0, 0);
    float acc = 0.0f;
    for (int t = 0; t < ntiles; ++t) {
        const int  b       = t & 1;
        const bool hasNext = (t + 1) < ntiles;
        if (hasNext) issue_tile(t + 1, (t + 1) & 1);   // overlap copy with compute
#if USE_ASYNC
        if (hasNext) wait_async_le6(); else wait_async_le0();  // tile t complete
#endif
        __syncthreads();                                // all waves' data visible
        const int base = t * TILE;
#pragma unroll 8
        for (int i = 0; i < TILE / 32; ++i) {
            const int kl = i * 32 + lane;
            const int k  = base + kl;
            const float cc = sC[b][kl];
            const float Pp = sP[b][kl];
            const float gg = sG[b][kl];
            const float arg  = fmaf(cc, S1s - Pp, 1.0f);          // 1 + c_k*(S1_s - P_k)
            const float pw   = fast_exp2(-beta * __log2f(arg));   // arg^-beta, raw trans
            const float term = gg * (1.0f - pw);
            const bool  ok   = (k >= 1) && (k <= step_s);
            acc += ok ? term : 0.0f;                               // select kills masked NaNs
        }
        __syncthreads();                                // buffer reusable next iter
    }

    // wave32 butterfly reduction
#pragma unroll
    for (int off = 16; off > 0; off >>= 1)
        acc += __shfl_xor(acc, off, 32);

    if (lane == 0 && s < M) {
        const float L0 = pL0[0], A = pA[0], alpha = pAlpha[0], B = pB[0];
        float pred = L0 + A * exp2f(-alpha * __log2f(S1s)) + B * acc;
        pred = fmaxf(pred, 1e-10f);
        const float r = __logf(loss[s]) - __logf(pred);
        const float a = fabsf(r);
        rowH[s] = (a <= HUB_DELTA) ? 0.5f * r * r
                                   : HUB_DELTA * (a - 0.5f * HUB_DELTA);
    }
}

// ---- Kernel 3: deterministic fixed-schedule reduction of M row values
__global__ __launch_bounds__(256)
void mpl_reduce(const float* __restrict__ rowH, float* __restrict__ out, int M) {
    __shared__ float sm[256];
    const int tid = threadIdx.x;
    float a = 0.0f;
    for (int i = tid; i < M; i += 256) a += rowH[i];   // fixed per-thread order
    sm[tid] = a;
    __syncthreads();
    for (int w = 128; w > 0; w >>= 1) {
        if (tid < w) sm[tid] += sm[tid + w];
        __syncthreads();
    }
    if (tid == 0) out[0] = sm[0];
}

extern "C" void kernel_launch(void* const* d_in, const int* in_sizes, int n_in,
                              void* d_out, int out_size, void* d_ws, size_t ws_size,
                              hipStream_t stream) {
    const float* S1     = (const float*)d_in[0];
    const float* lrs    = (const float*)d_in[1];
    const float* lr_sum = (const float*)d_in[2];
    const int*   step   = (const int*)  d_in[3];
    const float* lr_gap = (const float*)d_in[4];
    const float* loss   = (const float*)d_in[5];
    const float* pL0    = (const float*)d_in[6];
    const float* pA     = (const float*)d_in[7];
    const float* pAlpha = (const float*)d_in[8];
    const float* pB     = (const float*)d_in[9];
    const float* pC     = (const float*)d_in[10];
    const float* pBeta  = (const float*)d_in[11];
    const float* pGamma = (const float*)d_in[12];
    const int M = in_sizes[0];
    const int T = in_sizes[1];

    float* cArr = (float*)d_ws;        // T floats: C * lrs^-gamma
    float* pArr = cArr + T;            // T floats: lr_sum[k-1]
    float* rowH = pArr + T;            // M floats: per-row huber

    mpl_precompute<<<(T + 255) / 256, 256, 0, stream>>>(lrs, lr_sum, pC, pGamma,
                                                        cArr, pArr, T);
    const int blocks = (M + RPB - 1) / RPB;
    mpl_main_kernel<<<blocks, TPB, 0, stream>>>(S1, lr_gap, step, loss, cArr, pArr,
                                                pL0, pA, pAlpha, pB, pBeta,
                                                rowH, M, T);
    mpl_reduce<<<1, 256, 0, stream>>>(rowH, (float*)d_out, M);
}